// GCN_19550691131664
// MI455X (gfx1250) — compile-verified
//
#include <hip/hip_runtime.h>
#include <hip/hip_bf16.h>

typedef float v2f __attribute__((ext_vector_type(2)));
typedef float v8f __attribute__((ext_vector_type(8)));

#define IN_CH 128
#define HID   16
#define OUT_C 64

// ---------------------------------------------------------------------------
// Degree / normalization kernels
// ---------------------------------------------------------------------------
__global__ void k_deg_init(float* __restrict__ deg, int n) {
    int i = blockIdx.x * blockDim.x + threadIdx.x;
    if (i < n) deg[i] = 1.0f;  // self loop contributes 1
}

__global__ void k_deg_count(const long long* __restrict__ dst, float* __restrict__ deg,
                            long long nE) {
    long long e = (long long)blockIdx.x * blockDim.x + threadIdx.x;
    if (e < nE) atomicAdd(&deg[dst[e]], 1.0f);
}

__global__ void k_disq(const float* __restrict__ deg, float* __restrict__ disq, int n) {
    int i = blockIdx.x * blockDim.x + threadIdx.x;
    if (i < n) disq[i] = rsqrtf(deg[i]);  // deg >= 1 always
}

// ---------------------------------------------------------------------------
// GEMM1: Y1[N,16] = X[N,128] @ W1[128,16]   (fp32 WMMA 16x16x4)
// One wave handles a 16-row tile; 32 WMMA steps over K=128.
// ---------------------------------------------------------------------------
__global__ __launch_bounds__(256) void k_gemm1(const float* __restrict__ X,
                                               const float* __restrict__ W1,
                                               float* __restrict__ Y1, int n) {
    const int lane = threadIdx.x & 31;
    const int wave = threadIdx.x >> 5;
    const int tile = blockIdx.x * 8 + wave;
    if (tile * 16 >= n) return;            // uniform per wave, EXEC stays all-1s
    const int row0 = tile * 16;
    const int m    = lane & 15;            // A: row in tile, B: column
    const int koff = (lane >> 4) * 2;      // 0 or 2

    const float* xrow = X + (size_t)(row0 + m) * IN_CH;

    v8f acc = {};
    #pragma unroll
    for (int k0 = 0; k0 < IN_CH; k0 += 4) {
        v2f a, b;
        a.x = xrow[k0 + koff];
        a.y = xrow[k0 + koff + 1];
        b.x = W1[(k0 + koff)     * HID + m];
        b.y = W1[(k0 + koff + 1) * HID + m];
        acc = __builtin_amdgcn_wmma_f32_16x16x4_f32(false, a, false, b,
                                                    (short)0, acc, false, false);
    }
    const int rh = (lane >> 4) * 8;        // C/D: VGPR v -> row v + 8*(lane>>4)
    #pragma unroll
    for (int v = 0; v < 8; ++v)
        Y1[(size_t)(row0 + v + rh) * HID + m] = acc[v];
}

// ---------------------------------------------------------------------------
// Self-loop init: AGG[i,c] = Y[i,c] * disq[i]^2   (= Y/deg)
// ---------------------------------------------------------------------------
__global__ void k_self_init(const float* __restrict__ Y, const float* __restrict__ disq,
                            float* __restrict__ AGG, int n) {
    int i = blockIdx.x * blockDim.x + threadIdx.x;   // over n*HID
    if (i >= n * HID) return;
    float dq = disq[i >> 4];
    AGG[i] = Y[i] * dq * dq;
}

// ---------------------------------------------------------------------------
// Edge aggregation: 16 lanes per edge (lane = channel).
// AGG[dst,c] += disq[src]*disq[dst] * Y[src,c]
// ---------------------------------------------------------------------------
__global__ void k_edge_agg(const long long* __restrict__ src,
                           const long long* __restrict__ dst,
                           const float* __restrict__ disq,
                           const float* __restrict__ Y,
                           float* __restrict__ AGG, long long nE) {
    long long tid = (long long)blockIdx.x * blockDim.x + threadIdx.x;
    long long e = tid >> 4;
    int c = (int)(tid & 15);
    if (e >= nE) return;
    long long s = src[e];
    long long d = dst[e];
    float w = disq[s] * disq[d];
    atomicAdd(&AGG[d * HID + c], w * Y[s * HID + c]);
}

// ---------------------------------------------------------------------------
// H = relu(AGG1 + b1); AGG2 = H * disq^2  (self-loop init for layer 2)
// ---------------------------------------------------------------------------
__global__ void k_relu_init2(const float* __restrict__ AGG1, const float* __restrict__ b1,
                             const float* __restrict__ disq,
                             float* __restrict__ H, float* __restrict__ AGG2, int n) {
    int i = blockIdx.x * blockDim.x + threadIdx.x;
    if (i >= n * HID) return;
    float h = AGG1[i] + b1[i & 15];
    h = fmaxf(h, 0.0f);
    H[i] = h;
    float dq = disq[i >> 4];
    AGG2[i] = h * dq * dq;
}

// ---------------------------------------------------------------------------
// GEMM2: OUT[N,64] = Z[N,16] @ W2[16,64] + b2   (fp32 WMMA 16x16x4)
// One wave: 16 rows x 64 cols = 4 accumulator tiles; A reused across tiles.
// ---------------------------------------------------------------------------
__global__ __launch_bounds__(256) void k_gemm2(const float* __restrict__ Z,
                                               const float* __restrict__ W2,
                                               const float* __restrict__ b2,
                                               float* __restrict__ OUT, int n) {
    const int lane = threadIdx.x & 31;
    const int wave = threadIdx.x >> 5;
    const int tile = blockIdx.x * 8 + wave;
    if (tile * 16 >= n) return;
    const int row0 = tile * 16;
    const int m    = lane & 15;
    const int koff = (lane >> 4) * 2;

    const float* zrow = Z + (size_t)(row0 + m) * HID;

    v8f acc0 = {}, acc1 = {}, acc2 = {}, acc3 = {};
    #pragma unroll
    for (int k0 = 0; k0 < HID; k0 += 4) {
        v2f a;
        a.x = zrow[k0 + koff];
        a.y = zrow[k0 + koff + 1];
        const float* w0 = W2 + (k0 + koff)     * OUT_C + m;
        const float* w1 = W2 + (k0 + koff + 1) * OUT_C + m;
        v2f b;
        b.x = w0[0];  b.y = w1[0];
        acc0 = __builtin_amdgcn_wmma_f32_16x16x4_f32(false, a, false, b, (short)0, acc0, false, false);
        b.x = w0[16]; b.y = w1[16];
        acc1 = __builtin_amdgcn_wmma_f32_16x16x4_f32(false, a, false, b, (short)0, acc1, false, false);
        b.x = w0[32]; b.y = w1[32];
        acc2 = __builtin_amdgcn_wmma_f32_16x16x4_f32(false, a, false, b, (short)0, acc2, false, false);
        b.x = w0[48]; b.y = w1[48];
        acc3 = __builtin_amdgcn_wmma_f32_16x16x4_f32(false, a, false, b, (short)0, acc3, false, false);
    }

    const int rh = (lane >> 4) * 8;
    v8f* accs[4] = {&acc0, &acc1, &acc2, &acc3};
    #pragma unroll
    for (int t = 0; t < 4; ++t) {
        float bias = b2[t * 16 + m];
        #pragma unroll
        for (int v = 0; v < 8; ++v)
            OUT[(size_t)(row0 + v + rh) * OUT_C + t * 16 + m] = (*accs[t])[v] + bias;
    }
}

// ---------------------------------------------------------------------------
// Launch
// ---------------------------------------------------------------------------
extern "C" void kernel_launch(void* const* d_in, const int* in_sizes, int n_in,
                              void* d_out, int out_size, void* d_ws, size_t ws_size,
                              hipStream_t stream) {
    const float*     x   = (const float*)d_in[0];
    const long long* ei  = (const long long*)d_in[1];
    const float*     W1  = (const float*)d_in[2];
    const float*     b1  = (const float*)d_in[3];
    const float*     W2  = (const float*)d_in[4];
    const float*     b2  = (const float*)d_in[5];
    float*           out = (float*)d_out;

    const int       N = in_sizes[0] / IN_CH;        // 100000 (multiple of 16)
    const long long E = (long long)in_sizes[1] / 2; // 3200000
    const long long* srcI = ei;
    const long long* dstI = ei + E;

    // workspace layout
    float* deg  = (float*)d_ws;          // N
    float* disq = deg  + N;              // N
    float* Y1   = disq + N;              // N*16
    float* AGG1 = Y1   + (size_t)N * HID;
    float* H    = AGG1 + (size_t)N * HID;
    float* AGG2 = H    + (size_t)N * HID;

    const int TB = 256;
    const int nTiles  = N / 16;
    const int gTiles  = (nTiles + 7) / 8;                 // 8 waves per block
    const int gN      = (N + TB - 1) / TB;
    const int gNH     = (N * HID + TB - 1) / TB;
    const int gE      = (int)((E + TB - 1) / TB);
    const int gE16    = (int)((E * 16 + TB - 1) / TB);

    // degrees + symmetric norm
    k_deg_init <<<gN, TB, 0, stream>>>(deg, N);
    k_deg_count<<<gE, TB, 0, stream>>>(dstI, deg, E);
    k_disq     <<<gN, TB, 0, stream>>>(deg, disq, N);

    // layer 1: linear (WMMA) -> aggregate
    k_gemm1    <<<gTiles, TB, 0, stream>>>(x, W1, Y1, N);
    k_self_init<<<gNH, TB, 0, stream>>>(Y1, disq, AGG1, N);
    k_edge_agg <<<gE16, TB, 0, stream>>>(srcI, dstI, disq, Y1, AGG1, E);

    // relu + layer-2 self-loop init (aggregate BEFORE W2: (A·H)·W2 == A·(H·W2))
    k_relu_init2<<<gNH, TB, 0, stream>>>(AGG1, b1, disq, H, AGG2, N);
    k_edge_agg  <<<gE16, TB, 0, stream>>>(srcI, dstI, disq, H, AGG2, E);

    // layer 2 linear (WMMA) + bias
    k_gemm2    <<<gTiles, TB, 0, stream>>>(AGG2, W2, b2, out, N);
}